// SA_Layer_8813272891483
// MI455X (gfx1250) — compile-verified
//
#include <hip/hip_runtime.h>
#include <hip/hip_bf16.h>
#include <cstdint>
#include <cstddef>

// Problem constants (match reference)
#define BB   8
#define PP   8192
#define CINF 64          // input feature channels
#define MM   (PP / 4)    // 2048 centers per batch
#define NS   32          // nsample (k neighbors)
#define NN   (BB * MM)   // 16384 total centers
#define OC2V 128
#define EPSV 1e-5f

typedef float v2f __attribute__((ext_vector_type(2)));
typedef float v8f __attribute__((ext_vector_type(8)));

// ---------------------------------------------------------------------------
// CDNA5 async global->LDS helpers (ASYNCcnt-tracked copies).
// ---------------------------------------------------------------------------
__device__ __forceinline__ uint32_t lds_off(const void* p) {
  // generic -> AS(3) addrspacecast, then ptrtoint: LDS byte offset
  return (uint32_t)(unsigned long long)(__attribute__((address_space(3))) const void*)p;
}

__device__ __forceinline__ void async_load_b128(uint32_t lds_byte, uint64_t gaddr) {
  asm volatile("global_load_async_to_lds_b128 %0, %1, off"
               :: "v"(lds_byte), "v"(gaddr) : "memory");
}

__device__ __forceinline__ void wait_async0() {
#if __has_builtin(__builtin_amdgcn_s_wait_asynccnt)
  __builtin_amdgcn_s_wait_asynccnt(0);
#else
  asm volatile("s_wait_asynccnt 0x0" ::: "memory");
#endif
}

// ---------------------------------------------------------------------------
// Kernel 0: zero the BN-stats workspace (must be re-zeroed every call).
// ---------------------------------------------------------------------------
__global__ void zero_stats_kernel(float* __restrict__ s) {
  int t = blockIdx.x * blockDim.x + threadIdx.x;
  if (t < 2048) s[t] = 0.0f;
}

// ---------------------------------------------------------------------------
// Kernel 1: gather center coordinates -> first output tensor (B, M, 3).
// ---------------------------------------------------------------------------
__global__ void centers_kernel(const float* __restrict__ xyz,
                               float* __restrict__ outc) {
  int t = blockIdx.x * blockDim.x + threadIdx.x;
  if (t >= NN) return;
  int b = t / MM, m = t % MM;
  int pc = (int)((double)m * (double)(PP - 1) / (double)(MM - 1));
  const float* src = xyz + ((size_t)b * PP + pc) * 3;
  float* dst = outc + (size_t)t * 3;
  dst[0] = src[0]; dst[1] = src[1]; dst[2] = src[2];
}

// ---------------------------------------------------------------------------
// Kernel 2: KNN (top-32 nearest of 8192) + gather into x0[n][68][32].
// One wave32 per center. Channel 67 is zero padding so GEMM K-steps are 4-wide.
// ---------------------------------------------------------------------------
__global__ __launch_bounds__(128) void knn_gather_kernel(
    const float* __restrict__ xyz, const float* __restrict__ feats,
    float* __restrict__ x0) {
  const int wave = blockIdx.x * (blockDim.x >> 5) + (threadIdx.x >> 5);
  const int lane = threadIdx.x & 31;
  if (wave >= NN) return;  // whole wave exits together

  const int b = wave / MM, m = wave % MM;
  const int pc = (int)((double)m * (double)(PP - 1) / (double)(MM - 1));
  const float* xb = xyz + (size_t)b * PP * 3;
  const float c0 = xb[pc * 3 + 0], c1 = xb[pc * 3 + 1], c2 = xb[pc * 3 + 2];
  const float cn2 = c0 * c0 + c1 * c1 + c2 * c2;

  float kd = __builtin_inff();     // per-lane kept distance (one of top-32)
  int   ki = 0;                    // per-lane kept point index
  float worst = __builtin_inff();  // conservative (>= true max of kept)

  for (int p0 = 0; p0 < PP; p0 += 32) {
    const int p = p0 + lane;
    if (p0 + 32 < PP)  // stream-prefetch next chunk (global_prefetch_b8)
      __builtin_prefetch(xb + (p + 32) * 3, 0, 0);
    const float v0 = xb[p * 3 + 0], v1 = xb[p * 3 + 1], v2 = xb[p * 3 + 2];
    const float d2 = cn2 + (v0 * v0 + v1 * v1 + v2 * v2)
                   - 2.0f * (c0 * v0 + c1 * v1 + c2 * v2);
    unsigned mk = (unsigned)__ballot(d2 < worst);
    while (mk) {
      const int src = __ffs(mk) - 1;
      mk &= mk - 1;
      const float cd = __shfl(d2, src, 32);
      const int ci = p0 + src;
      // argmax over kept entries (all 32 lanes get (mx, ml))
      float mx = kd; int ml = lane;
      #pragma unroll
      for (int off = 16; off; off >>= 1) {
        const float om = __shfl_xor(mx, off, 32);
        const int   ol = __shfl_xor(ml, off, 32);
        if (om > mx || (om == mx && ol < ml)) { mx = om; ml = ol; }
      }
      if (cd < mx && lane == ml) { kd = cd; ki = ci; }
      worst = mx;  // stale-by-one but conservative: never below true max
    }
  }

  // Gather: lane j owns neighbor slot j. x0 layout: [n][c(68)][j(32)].
  float* dst = x0 + (size_t)wave * 68 * NS;
  const float* fb = feats + (size_t)b * CINF * PP;
  const int idx = ki;
  dst[0 * NS + lane] = xb[idx * 3 + 0];
  dst[1 * NS + lane] = xb[idx * 3 + 1];
  dst[2 * NS + lane] = xb[idx * 3 + 2];
  #pragma unroll 4
  for (int c = 0; c < CINF; ++c)
    dst[(3 + c) * NS + lane] = fb[(size_t)c * PP + idx];
  dst[67 * NS + lane] = 0.0f;  // pad channel
}

// ---------------------------------------------------------------------------
// Kernel 3 (x3): GEMM y = W*x + b with
//  - async global->LDS double-buffered staging of activation panels
//    (each panel read from global exactly once per block, all o-tiles share it)
//  - fused pre-op (previous layer's BN+ReLU) applied on LDS load
//  - fused BN statistics (sum, sumsq) via half-wave reductions + f32 atomics
// V_WMMA_F32_16X16X4_F32: A = 16(o) x 4(c) weights, B = 4(c) x 16(j) acts.
// Wave w of 8: o-tile = w % (OC/16), panel-set = w / (OC/16).
// ---------------------------------------------------------------------------
template <int IC, int ICP, int KSTEPS, int OC, int NSTEPS>
__global__ __launch_bounds__(256) void gemm_bn_stats_kernel(
    const float* __restrict__ x, const float* __restrict__ W,
    const float* __restrict__ bias,
    const float* __restrict__ pre_scale, const float* __restrict__ pre_shift,
    float* __restrict__ y, float* __restrict__ osum,
    float* __restrict__ osumsq) {
  constexpr int OTILES = OC / 16;       // 4 or 8
  constexpr int NPS = 8 / OTILES;       // panels (n's) per step: 2 or 1
  constexpr int PANEL = ICP * NS;       // floats per panel
  constexpr int CHUNKS = (NPS * PANEL * 4) / 16;  // 16B chunks per step

  __shared__ __align__(16) float sx[2][NPS][PANEL];
  __shared__ float s_sc[ICP];
  __shared__ float s_sh[ICP];

  const bool preop = (pre_scale != nullptr);
  if (preop) {
    for (int c = threadIdx.x; c < ICP; c += blockDim.x) {
      s_sc[c] = (c < IC) ? pre_scale[c] : 1.0f;
      s_sh[c] = (c < IC) ? pre_shift[c] : 0.0f;
    }
  }

  const int lane = threadIdx.x & 31;
  const int lm = lane & 15;    // position within half
  const int hi = lane >> 4;    // which half of the wave
  const int w = threadIdx.x >> 5;
  const int otile = w % OTILES;
  const int pset = w / OTILES;
  const int blockBase = blockIdx.x * (NSTEPS * NPS);

  // Cooperative async staging of the NPS panels of step `sidx` into buf.
  auto stage = [&](int sidx, int buf) {
    const float* gsrc = x + (size_t)(blockBase + sidx * NPS) * PANEL;
    const uint32_t lbase = lds_off(&sx[buf][0][0]);
    const uint64_t gbase = (uint64_t)(uintptr_t)gsrc;
    for (int i = threadIdx.x; i < CHUNKS; i += 256)
      async_load_b128(lbase + (uint32_t)i * 16u, gbase + (uint64_t)i * 16u);
  };

  // A fragments (weights) resident in registers for all K-steps.
  const int orow = otile * 16 + lm;
  v2f A[KSTEPS];
  #pragma unroll
  for (int ks = 0; ks < KSTEPS; ++ks) {
    const int cA = ks * 4 + 2 * hi;
    const int cB = cA + 1;
    A[ks].x = (cA < IC) ? W[(size_t)orow * IC + cA] : 0.0f;
    A[ks].y = (cB < IC) ? W[(size_t)orow * IC + cB] : 0.0f;
  }
  float bv[8];
  #pragma unroll
  for (int r = 0; r < 8; ++r) bv[r] = bias[otile * 16 + r + 8 * hi];

  float accS[8], accQ[8];
  #pragma unroll
  for (int r = 0; r < 8; ++r) { accS[r] = 0.0f; accQ[r] = 0.0f; }

  stage(0, 0);  // prologue

  for (int s = 0; s < NSTEPS; ++s) {
    wait_async0();      // this wave's issued copies done
    __syncthreads();    // => buffer s&1 fully populated by all waves
    const int buf = s & 1;
    if (s + 1 < NSTEPS) stage(s + 1, buf ^ 1);  // overlap next copy

    const int n = blockBase + s * NPS + pset;
    const float* panel = &sx[buf][pset][0];
    #pragma unroll
    for (int jh = 0; jh < 2; ++jh) {
      const int j = jh * 16 + lm;
      v8f C = {0.f, 0.f, 0.f, 0.f, 0.f, 0.f, 0.f, 0.f};
      #pragma unroll
      for (int ks = 0; ks < KSTEPS; ++ks) {
        const int cA = ks * 4 + 2 * hi;
        const int cB = cA + 1;
        float bx = panel[cA * NS + j];   // ds_load
        float by = panel[cB * NS + j];
        if (preop) {  // previous layer's BN + ReLU, fused on load
          bx = fmaxf(fmaf(bx, s_sc[cA], s_sh[cA]), 0.0f);
          by = fmaxf(fmaf(by, s_sc[cB], s_sh[cB]), 0.0f);
        }
        v2f Bv; Bv.x = bx; Bv.y = by;
        C = __builtin_amdgcn_wmma_f32_16x16x4_f32(false, A[ks], false, Bv,
                                                  (short)0, C, false, false);
      }
      float* yb = y + ((size_t)n * OC + otile * 16 + 8 * hi) * NS + j;
      #pragma unroll
      for (int r = 0; r < 8; ++r) {
        const float yv = C[r] + bv[r];
        accS[r] += yv;
        accQ[r] += yv * yv;
        yb[(size_t)r * NS] = yv;
      }
    }
    __syncthreads();  // done reading buf before it is restaged at step s+2
  }

  // Reduce stats within each 16-lane half (rows o and o+8), then atomics.
  #pragma unroll
  for (int r = 0; r < 8; ++r) {
    float sv = accS[r], q = accQ[r];
    #pragma unroll
    for (int off = 8; off; off >>= 1) {
      sv += __shfl_xor(sv, off, 16);
      q += __shfl_xor(q, off, 16);
    }
    if (lm == 0) {
      const int o = otile * 16 + r + 8 * hi;
      atomicAdd(&osum[o], sv);
      atomicAdd(&osumsq[o], q);
    }
  }
}

// ---------------------------------------------------------------------------
// Kernel 4 (x3): finalize BN stats -> per-channel (scale, shift).
// ---------------------------------------------------------------------------
__global__ void finalize_stats_kernel(const float* __restrict__ sum,
                                      const float* __restrict__ sumsq,
                                      const float* __restrict__ g,
                                      const float* __restrict__ be,
                                      float* __restrict__ scale,
                                      float* __restrict__ shift, int OC,
                                      float invCnt) {
  int o = blockIdx.x * blockDim.x + threadIdx.x;
  if (o >= OC) return;
  const float mu = sum[o] * invCnt;
  const float var = sumsq[o] * invCnt - mu * mu;
  const float sc = g[o] * rsqrtf(var + EPSV);
  scale[o] = sc;
  shift[o] = be[o] - mu * sc;
}

// ---------------------------------------------------------------------------
// Kernel 5: last layer BN + ReLU + max over k, write (B, 128, M).
// ---------------------------------------------------------------------------
__global__ void maxpool_kernel(const float* __restrict__ y2,
                               const float* __restrict__ scale,
                               const float* __restrict__ shift,
                               float* __restrict__ out) {
  int t = blockIdx.x * blockDim.x + threadIdx.x;
  if (t >= NN * OC2V) return;
  const int n = t / OC2V, o = t % OC2V;
  const float sc = scale[o], sh = shift[o];
  const float4* src = (const float4*)(y2 + ((size_t)n * OC2V + o) * NS);
  float mx = -__builtin_inff();
  #pragma unroll
  for (int i = 0; i < 8; ++i) {
    float4 v = src[i];
    v.x = fmaxf(fmaf(v.x, sc, sh), 0.0f);
    v.y = fmaxf(fmaf(v.y, sc, sh), 0.0f);
    v.z = fmaxf(fmaf(v.z, sc, sh), 0.0f);
    v.w = fmaxf(fmaf(v.w, sc, sh), 0.0f);
    mx = fmaxf(mx, fmaxf(fmaxf(v.x, v.y), fmaxf(v.z, v.w)));
  }
  const int b = n / MM, m = n % MM;
  out[(size_t)b * OC2V * MM + (size_t)o * MM + m] = mx;
}

// ---------------------------------------------------------------------------
extern "C" void kernel_launch(void* const* d_in, const int* in_sizes, int n_in,
                              void* d_out, int out_size, void* d_ws,
                              size_t ws_size, hipStream_t stream) {
  (void)in_sizes; (void)n_in; (void)out_size; (void)ws_size;
  const float* xyz   = (const float*)d_in[0];
  const float* feats = (const float*)d_in[1];
  const float* W0 = (const float*)d_in[2];
  const float* b0 = (const float*)d_in[3];
  const float* g0 = (const float*)d_in[4];
  const float* be0 = (const float*)d_in[5];
  const float* W1 = (const float*)d_in[6];
  const float* b1 = (const float*)d_in[7];
  const float* g1 = (const float*)d_in[8];
  const float* be1 = (const float*)d_in[9];
  const float* W2 = (const float*)d_in[10];
  const float* b2 = (const float*)d_in[11];
  const float* g2 = (const float*)d_in[12];
  const float* be2 = (const float*)d_in[13];

  float* out = (float*)d_out;  // centers (B*M*3) then (B,128,M)
  float* ws = (float*)d_ws;

  // Workspace layout (floats)
  float* stats = ws;  // 2048 floats: per layer l: sum, sumsq, scale, shift
  float* sum0 = stats + 0 * 512; float* ssq0 = sum0 + 128;
  float* sc0  = sum0 + 256;      float* sh0  = sum0 + 384;
  float* sum1 = stats + 1 * 512; float* ssq1 = sum1 + 128;
  float* sc1  = sum1 + 256;      float* sh1  = sum1 + 384;
  float* sum2 = stats + 2 * 512; float* ssq2 = sum2 + 128;
  float* sc2  = sum2 + 256;      float* sh2  = sum2 + 384;

  float* x0 = ws + 2048;                  // NN*68*32
  float* y0 = x0 + (size_t)NN * 68 * NS;  // NN*64*32
  float* y1 = x0;                         // alias: x0 dead after L0
  float* y2 = y0 + (size_t)NN * 64 * NS;  // NN*128*32

  const float invCnt = 1.0f / (float)((size_t)NN * NS);

  zero_stats_kernel<<<8, 256, 0, stream>>>(stats);
  centers_kernel<<<(NN + 255) / 256, 256, 0, stream>>>(xyz, out);
  knn_gather_kernel<<<NN / 4, 128, 0, stream>>>(xyz, feats, x0);

  // Layer 0: 67 -> 64 (identity pre-op).  NPS=2, NSTEPS=16 -> 512 blocks.
  gemm_bn_stats_kernel<67, 68, 17, 64, 16><<<512, 256, 0, stream>>>(
      x0, W0, b0, nullptr, nullptr, y0, sum0, ssq0);
  finalize_stats_kernel<<<1, 64, 0, stream>>>(sum0, ssq0, g0, be0, sc0, sh0, 64,
                                              invCnt);
  // Layer 1: 64 -> 64 (pre-op = BN0 + ReLU)
  gemm_bn_stats_kernel<64, 64, 16, 64, 16><<<512, 256, 0, stream>>>(
      y0, W1, b1, sc0, sh0, y1, sum1, ssq1);
  finalize_stats_kernel<<<1, 64, 0, stream>>>(sum1, ssq1, g1, be1, sc1, sh1, 64,
                                              invCnt);
  // Layer 2: 64 -> 128 (pre-op = BN1 + ReLU).  NPS=1, NSTEPS=32 -> 512 blocks.
  gemm_bn_stats_kernel<64, 64, 16, 128, 32><<<512, 256, 0, stream>>>(
      y1, W2, b2, sc1, sh1, y2, sum2, ssq2);
  finalize_stats_kernel<<<1, 128, 0, stream>>>(sum2, ssq2, g2, be2, sc2, sh2,
                                               128, invCnt);
  // BN2 + ReLU + max over k -> output
  maxpool_kernel<<<(NN * OC2V + 255) / 256, 256, 0, stream>>>(
      y2, sc2, sh2, out + (size_t)NN * 3);
}